// LTIExact_9113920602491
// MI455X (gfx1250) — compile-verified
//
#include <hip/hip_runtime.h>
#include <hip/hip_bf16.h>
#include <stdint.h>

typedef __attribute__((ext_vector_type(2))) float v2f;
typedef __attribute__((ext_vector_type(8))) float v8f;

#define DDIM 64
#define TDIM 256
#define BDIM 4096
#define LDA  68   // LDS row stride (floats): conflict-free AND 16B-aligned rows

// ---------------------------------------------------------------------------
// Kernel 1: prep
//   A  = W * mask                     (64x64)
//   Xc = X0 - x_star (broadcast row)  (4096x64)
// ---------------------------------------------------------------------------
__global__ void lti_prep(const float* __restrict__ W,
                         const unsigned char* __restrict__ mask,
                         const float* __restrict__ X0,
                         const float* __restrict__ xstar,
                         float* __restrict__ A,
                         float* __restrict__ Xc) {
    int i = blockIdx.x * blockDim.x + threadIdx.x;
    if (i < DDIM * DDIM) {
        A[i] = W[i] * (mask[i] ? 1.0f : 0.0f);
    }
    if (i < BDIM * DDIM) {
        Xc[i] = X0[i] - xstar[i & (DDIM - 1)];
    }
}

// ---------------------------------------------------------------------------
// 64x64 fp32 matmul in LDS using V_WMMA_F32_16X16X4_F32.
// 8 waves per block; wave w computes tiles 2w, 2w+1 of the 4x4 tile grid.
// lC = lA @ lB.  Caller handles barriers.
// ---------------------------------------------------------------------------
__device__ __forceinline__ void mm64_lds(const float* __restrict__ lA,
                                         const float* __restrict__ lB,
                                         float* __restrict__ lC) {
    const int lane = threadIdx.x & 31;
    const int wave = threadIdx.x >> 5;
    const int half = lane >> 4;      // 0: K pair {0,1}, 1: K pair {2,3}
    const int l    = lane & 15;

    #pragma unroll
    for (int tt = 0; tt < 2; ++tt) {
        const int tile = wave * 2 + tt;
        const int ti = tile >> 2;
        const int tj = tile & 3;
        v8f acc = {};
        #pragma unroll
        for (int kc = 0; kc < 16; ++kc) {
            const int k0 = kc * 4 + half * 2;
            v2f a = *(const v2f*)(lA + (ti * 16 + l) * LDA + k0);  // 8B aligned
            v2f b;
            b.x = lB[(k0) * LDA + tj * 16 + l];
            b.y = lB[(k0 + 1) * LDA + tj * 16 + l];
            acc = __builtin_amdgcn_wmma_f32_16x16x4_f32(
                false, a, false, b, (short)0, acc, false, false);
        }
        #pragma unroll
        for (int r = 0; r < 8; ++r) {
            lC[(ti * 16 + r + half * 8) * LDA + tj * 16 + l] = acc[r];
        }
    }
}

// ---------------------------------------------------------------------------
// Kernel 2: batched matrix exponential, one workgroup per time point.
//   Bm = A * t / 2^8 ; Taylor order 8 via Horner ; square 8 times.
// ---------------------------------------------------------------------------
__global__ void lti_expm(const float* __restrict__ A,
                         const float* __restrict__ t_span,
                         float* __restrict__ E) {
    __shared__ float lB[DDIM * LDA];
    __shared__ float lR[DDIM * LDA];
    __shared__ float lT[DDIM * LDA];

    const int t   = blockIdx.x;
    const int tid = threadIdx.x;
    const float sc = t_span[t] * (1.0f / 256.0f);   // t / 2^8

    for (int i = tid; i < DDIM * DDIM; i += 256) {
        const int r = i >> 6, c = i & 63;
        lB[r * LDA + c] = A[i] * sc;
        lR[r * LDA + c] = (r == c) ? 1.0f : 0.0f;
    }
    __syncthreads();

    // Horner: R = I + (Bm @ R) / j,  j = 8..1  (unrolled so 1/j is a constant)
    #pragma unroll
    for (int j = 8; j >= 1; --j) {
        mm64_lds(lB, lR, lT);
        __syncthreads();
        const float inv = 1.0f / (float)j;
        for (int i = tid; i < DDIM * DDIM; i += 256) {
            const int r = i >> 6, c = i & 63;
            lR[r * LDA + c] = ((r == c) ? 1.0f : 0.0f) + lT[r * LDA + c] * inv;
        }
        __syncthreads();
    }

    // square 8 times: R = R @ R
    for (int s = 0; s < 8; ++s) {
        mm64_lds(lR, lR, lT);
        __syncthreads();
        for (int i = tid; i < DDIM * DDIM; i += 256) {
            lR[(i >> 6) * LDA + (i & 63)] = lT[(i >> 6) * LDA + (i & 63)];
        }
        __syncthreads();
    }

    float* __restrict__ Et = E + (size_t)t * DDIM * DDIM;
    for (int i = tid; i < DDIM * DDIM; i += 256) {
        Et[i] = lR[(i >> 6) * LDA + (i & 63)];
    }
}

// ---------------------------------------------------------------------------
// Kernel 3: main GEMM.  grid = (B/256, T); 8 waves/block.
//   out[b, t, k] = sum_d Xc[b, d] * E[t, k, d] + x_star[k]
// Each wave: 32 rows (two 16-row M-tiles sharing B fragments) x 64 columns.
// E[t] staged into LDS via async DMA (global_load_async_to_lds_b128).
// ---------------------------------------------------------------------------
__global__ void lti_gemm(const float* __restrict__ Xc,
                         const float* __restrict__ E,
                         const float* __restrict__ xstar,
                         float* __restrict__ out) {
    __shared__ float lE[DDIM * LDA];

    const int t   = blockIdx.y;
    const int tid = threadIdx.x;

    // ---- async stage E[t] (16 KB) into LDS, padded stride ----
    {
        const float* __restrict__ Et = E + (size_t)t * DDIM * DDIM;
        const unsigned ldsBase = (unsigned)(uintptr_t)lE;  // low 32 bits = LDS offset
        #pragma unroll
        for (int it = 0; it < 4; ++it) {
            const int c   = tid + it * 256;       // 16B chunk id, 0..1023
            const int row = c >> 4;
            const int col = (c & 15) * 4;
            const unsigned ldsOff = ldsBase + (unsigned)((row * LDA + col) * 4);
            const float* g = Et + c * 4;
            asm volatile("global_load_async_to_lds_b128 %0, %1, off"
                         :: "v"(ldsOff), "v"(g) : "memory");
        }
        asm volatile("s_wait_asynccnt 0" ::: "memory");
    }
    __syncthreads();

    const int lane = tid & 31;
    const int wave = tid >> 5;
    const int half = lane >> 4;
    const int l    = lane & 15;
    const int b0   = blockIdx.x * 256 + wave * 32;   // two 16-row tiles: b0, b0+16

    // preload A fragments for both M-tiles (16 v2f each)
    v2f av0[16], av1[16];
    {
        const float* __restrict__ xr0 = Xc + (size_t)(b0 + l) * DDIM;
        const float* __restrict__ xr1 = Xc + (size_t)(b0 + 16 + l) * DDIM;
        #pragma unroll
        for (int kc = 0; kc < 16; ++kc) {
            const int k0 = kc * 4 + half * 2;
            av0[kc] = *(const v2f*)(xr0 + k0);
            av1[kc] = *(const v2f*)(xr1 + k0);
        }
    }

    #pragma unroll
    for (int n = 0; n < 4; ++n) {
        // B fragments for this N-tile, shared by both M-tiles
        v2f bv[16];
        #pragma unroll
        for (int kc = 0; kc < 16; ++kc) {
            const int k0 = kc * 4 + half * 2;
            bv[kc] = *(const v2f*)(lE + (n * 16 + l) * LDA + k0);  // ds_load_b64
        }

        const float xsv = xstar[n * 16 + l];
        v8f acc0 = {xsv, xsv, xsv, xsv, xsv, xsv, xsv, xsv};
        v8f acc1 = acc0;
        #pragma unroll
        for (int kc = 0; kc < 16; ++kc) {
            acc0 = __builtin_amdgcn_wmma_f32_16x16x4_f32(
                false, av0[kc], false, bv[kc], (short)0, acc0, false, false);
            acc1 = __builtin_amdgcn_wmma_f32_16x16x4_f32(
                false, av1[kc], false, bv[kc], (short)0, acc1, false, false);
        }

        #pragma unroll
        for (int r = 0; r < 8; ++r) {
            const int M = r + half * 8;
            out[((size_t)(b0 + M) * TDIM + t) * DDIM + n * 16 + l] = acc0[r];
            out[((size_t)(b0 + 16 + M) * TDIM + t) * DDIM + n * 16 + l] = acc1[r];
        }
    }
}

// ---------------------------------------------------------------------------
// launch
// ---------------------------------------------------------------------------
extern "C" void kernel_launch(void* const* d_in, const int* in_sizes, int n_in,
                              void* d_out, int out_size, void* d_ws, size_t ws_size,
                              hipStream_t stream) {
    const float*         X0    = (const float*)d_in[0];
    const float*         tspan = (const float*)d_in[1];
    const float*         W     = (const float*)d_in[2];
    const float*         xstar = (const float*)d_in[3];
    const unsigned char* mask  = (const unsigned char*)d_in[4];

    float* out = (float*)d_out;

    // workspace layout (floats): A[4096] | Xc[4096*64] | E[256*64*64]
    float* A  = (float*)d_ws;
    float* Xc = A + DDIM * DDIM;
    float* E  = Xc + (size_t)BDIM * DDIM;

    {
        const int total = BDIM * DDIM;
        lti_prep<<<(total + 255) / 256, 256, 0, stream>>>(W, mask, X0, xstar, A, Xc);
    }
    lti_expm<<<TDIM, 256, 0, stream>>>(A, tspan, E);
    lti_gemm<<<dim3(BDIM / 256, TDIM), 256, 0, stream>>>(Xc, E, xstar, out);
}